// InterpolatingBSpline3d_8735963480090
// MI455X (gfx1250) — compile-verified
//
#include <hip/hip_runtime.h>
#include <hip/hip_bf16.h>

typedef __attribute__((ext_vector_type(2))) float v2f;
typedef __attribute__((ext_vector_type(8))) float v8f;

#define MM 66          // M + 2, M = 64
#define CH_STRIDE (66*66*66)   // 287496

// ---------------------------------------------------------------------------
// Kernel 1: build the fixed 66x66 spline system matrix A and invert it with
// Gauss-Jordan in LDS (single workgroup).  Writes W[t][i] = invA[i][t+1]
// (64 x 66, row-major) so each solve pass becomes  out = rhs @ W.
// ---------------------------------------------------------------------------
__global__ void spline_invert_kernel(float* __restrict__ Wout) {
    __shared__ float A[MM][MM];
    __shared__ float Inv[MM][MM];
    __shared__ float fac[MM];
    const int tid = threadIdx.x;
    const int nt  = blockDim.x;
    const float dis = 63.0f * 63.0f;   // (1/delta)^2, delta = 1/(M-1)

    for (int idx = tid; idx < MM * MM; idx += nt) {
        int r = idx / MM, c = idx % MM;
        float a = 0.0f;
        if (r == 0) {
            if (c == 0) a = dis; else if (c == 1) a = -2.0f * dis; else if (c == 2) a = dis;
        } else if (r == MM - 1) {
            if (c == MM - 3) a = dis; else if (c == MM - 2) a = -2.0f * dis; else if (c == MM - 1) a = dis;
        } else {
            if (c == r - 1) a = 1.0f / 6.0f;
            else if (c == r) a = 2.0f / 3.0f;
            else if (c == r + 1) a = 1.0f / 6.0f;
        }
        A[r][c] = a;
        Inv[r][c] = (r == c) ? 1.0f : 0.0f;
    }
    __syncthreads();

    for (int k = 0; k < MM; ++k) {
        if (tid < MM) fac[tid] = A[tid][k];
        __syncthreads();
        float piv = 1.0f / fac[k];
        for (int c = tid; c < MM; c += nt) { A[k][c] *= piv; Inv[k][c] *= piv; }
        __syncthreads();
        for (int idx = tid; idx < MM * MM; idx += nt) {
            int r = idx / MM, c = idx % MM;
            if (r != k) {
                float f = fac[r];
                A[r][c]   -= f * A[k][c];
                Inv[r][c] -= f * Inv[k][c];
            }
        }
        __syncthreads();
    }

    // W[t][i] = invA[i][t+1]  (the zero-padded RHS drops columns 0 and 65)
    for (int idx = tid; idx < 64 * MM; idx += nt) {
        int t = idx / MM, i = idx % MM;
        Wout[t * MM + i] = Inv[i][t + 1];
    }
}

// ---------------------------------------------------------------------------
// Kernel 2: strided WMMA-f32 GEMM, all strides compile-time.
//   out[r, k] = sum_t in(r, t) * W[t, k],  t in [0,64), k in [0,66)
//   in(r, t)  = in [(r >> L)*OSIN  + (r & ((1<<L)-1)) + t*TSTRIDE]
//   out(r, k) = out[(r >> L)*OSOUT + (r & ((1<<L)-1)) + k*KSTRIDE]
// One wave (32 threads) computes one 16x16 (r, k) tile with
// v_wmma_f32_16x16x4_f32 over 16 K-steps.  5 column tiles cover k<66.
//
// f32 A 16x4 layout:  lane = 16*h + m  holds A[m, kb+2h] (v0), A[m, kb+2h+1] (v1)
// f32 B 4x16 layout:  lane = 16*h + n  holds B[kb+2h, n] (v0), B[kb+2h+1, n] (v1)
// f32 C/D 16x16:      vgpr v, lane = 16*h + n  holds D[v + 8h, n]
// ---------------------------------------------------------------------------
template <int L, int OSIN, int TSTRIDE, int OSOUT, int KSTRIDE, bool GUARD_R>
__global__ void solve_gemm_kernel(const float* __restrict__ in,
                                  const float* __restrict__ W,
                                  float* __restrict__ out, int N) {
    constexpr int IMASK = (1 << L) - 1;

    const int tile  = blockIdx.x;
    const int rTile = tile / 5;
    const int kTile = tile % 5;
    const int r0 = rTile * 16;
    const int k0 = kTile * 16;

    const int lane = threadIdx.x;      // 0..31, single wave
    const int half = lane >> 4;        // 0 or 1
    const int m    = lane & 15;

    // A-fragment row (clamped; junk lanes masked at store time)
    int r  = r0 + m;
    int rc = (r < N) ? r : (N - 1);
    const int inBase = (rc >> L) * OSIN + (rc & IMASK);

    // B-fragment column (clamped; junk cols masked at store time)
    int kcol = k0 + m;
    int kc   = (kcol < MM) ? kcol : (MM - 1);

    v8f acc = {};
    for (int kb = 0; kb < 64; kb += 4) {
        const int t0 = kb + 2 * half;
        v2f a, b;
        a.x = in[inBase + t0 * TSTRIDE];
        a.y = in[inBase + (t0 + 1) * TSTRIDE];
        b.x = W[t0 * MM + kc];
        b.y = W[(t0 + 1) * MM + kc];
        acc = __builtin_amdgcn_wmma_f32_16x16x4_f32(
                /*neg_a=*/false, a, /*neg_b=*/false, b,
                /*c_mod=*/(short)0, acc, /*reuse_a=*/false, /*reuse_b=*/false);
    }

    // Store: this lane owns rows r0 + 8*half + {0..7}, column k0 + m.
    const int col = k0 + m;
    const bool colOk = (col < MM);
    const int rowBase = r0 + 8 * half;
    #pragma unroll
    for (int v = 0; v < 8; ++v) {
        int row = rowBase + v;
        bool ok = colOk && (!GUARD_R || row < N);
        if (ok) {
            int oa = (row >> L) * OSOUT + (row & IMASK) + col * KSTRIDE;
            out[oa] = acc[v];
        }
    }
}

// ---------------------------------------------------------------------------
// Kernel 3: evaluate the cubic B-spline at 1M points, 2 channels.
// coefs layout: (2, 66, 66, 66) row-major.  Output: (P, 2).
// ---------------------------------------------------------------------------
__device__ __forceinline__ void idx_frac(float un, int M, int& i, float& t) {
    float fi = floorf(un);
    i = (int)fi;
    t = un - fi;
    if (un < 0.0f)            { i = 0;     t = un; }
    if (un >= (float)(M - 2)) { i = M - 2; t = un - (float)(M - 2); }
}

__device__ __forceinline__ void bspline_basis(float t, float b[4]) {
    float t2 = t * t, t3 = t2 * t;
    b[0] = (1.0f / 6.0f) * (-t3 + 3.0f * t2 - 3.0f * t + 1.0f);
    b[1] = (1.0f / 6.0f) * (3.0f * t3 - 6.0f * t2 + 4.0f);
    b[2] = (1.0f / 6.0f) * (-3.0f * t3 + 3.0f * t2 + 3.0f * t + 1.0f);
    b[3] = (1.0f / 6.0f) * t3;
}

__global__ void interp_kernel(const float* __restrict__ u,
                              const float* __restrict__ coefs,
                              float* __restrict__ out, int P) {
    int p = blockIdx.x * blockDim.x + threadIdx.x;
    if (p >= P) return;

    float ux = u[3 * p + 0], uy = u[3 * p + 1], uz = u[3 * p + 2];
    int ix, iy, iz;
    float tx, ty, tz;
    idx_frac(ux * 63.0f, 64, ix, tx);
    idx_frac(uy * 63.0f, 64, iy, ty);
    idx_frac(uz * 63.0f, 64, iz, tz);

    float bx[4], by[4], bz[4];
    bspline_basis(tx, bx);
    bspline_basis(ty, by);
    bspline_basis(tz, bz);

    const float* C0 = coefs;
    const float* C1 = coefs + CH_STRIDE;

    float a0 = 0.0f, a1 = 0.0f;
    #pragma unroll
    for (int di = 0; di < 4; ++di) {
        #pragma unroll
        for (int dj = 0; dj < 4; ++dj) {
            float w = bx[di] * by[dj];
            int base = ((ix + di) * MM + (iy + dj)) * MM + iz;
            float s0 = C0[base + 0] * bz[0] + C0[base + 1] * bz[1]
                     + C0[base + 2] * bz[2] + C0[base + 3] * bz[3];
            float s1 = C1[base + 0] * bz[0] + C1[base + 1] * bz[1]
                     + C1[base + 2] * bz[2] + C1[base + 3] * bz[3];
            a0 = fmaf(w, s0, a0);
            a1 = fmaf(w, s1, a1);
        }
    }
    out[2 * p + 0] = a0;
    out[2 * p + 1] = a1;
}

// ---------------------------------------------------------------------------
extern "C" void kernel_launch(void* const* d_in, const int* in_sizes, int n_in,
                              void* d_out, int out_size, void* d_ws, size_t ws_size,
                              hipStream_t stream) {
    const float* u    = (const float*)d_in[0];   // (1000000, 3) f32
    const float* data = (const float*)d_in[1];   // (2, 64, 64, 64) f32
    float* out = (float*)d_out;                  // (1000000, 2) f32
    const int P = in_sizes[0] / 3;

    // workspace layout (floats)
    float* ws    = (float*)d_ws;
    float* W     = ws;                     // 64*66            = 4224
    float* cx    = W   + 64 * MM;          // 2*66*64*64       = 540672
    float* cxy   = cx  + 2 * MM * 64 * 64; // 2*66*66*64       = 557568
    float* coefs = cxy + 2 * MM * MM * 64; // 2*66*66*66       = 574992

    // 1) invert the fixed system matrix
    spline_invert_kernel<<<1, 256, 0, stream>>>(W);

    // 2) solve along x:  cx[c,i,y,z] = sum_x W[x,i] * data[c,x,y,z]
    //    r = c*4096 + (y*64+z), N = 8192 (multiple of 16 -> no row guard)
    solve_gemm_kernel<12, 64*64*64, 4096, MM*4096, 4096, false>
        <<<(8192 / 16) * 5, 32, 0, stream>>>(data, W, cx, 8192);

    // 3) solve along y:  cxy[c,i,j,z] = sum_y W[y,j] * cx[c,i,y,z]
    //    r = (c*66+i)*64 + z, N = 8448 (multiple of 16 -> no row guard)
    solve_gemm_kernel<6, 4096, 64, MM*64, 64, false>
        <<<(8448 / 16) * 5, 32, 0, stream>>>(cx, W, cxy, 8448);

    // 4) solve along z:  coefs[c,i,j,k] = sum_z W[z,k] * cxy[c,i,j,z]
    //    r = (c*66+i)*66 + j, N = 8712 (ragged -> row guard on stores)
    solve_gemm_kernel<0, 64, 1, MM, 1, true>
        <<<((8712 + 15) / 16) * 5, 32, 0, stream>>>(cxy, W, coefs, 8712);

    // 5) interpolate 1M points
    interp_kernel<<<(P + 255) / 256, 256, 0, stream>>>(u, coefs, out, P);
}